// RGCN_29145648070962
// MI455X (gfx1250) — compile-verified
//
#include <hip/hip_runtime.h>
#include <hip/hip_bf16.h>

typedef __attribute__((ext_vector_type(2))) float v2f;
typedef __attribute__((ext_vector_type(8))) float v8f;

#define TILE_K 64

__device__ __forceinline__ v8f wmma_f32(v2f a, v2f b, v8f c) {
    return __builtin_amdgcn_wmma_f32_16x16x4_f32(false, a, false, b, (short)0, c, false, false);
}

// C[M x N] = A[M x K] * B[K x N] (+ bias[N] if bias != nullptr)
// Block: 256 threads = 8 waves (wave32). Each wave computes a 32x64 tile
// (2 A-fragments x 4 B-fragments = 8 v8f accumulators).
// Block tile: 256 rows x 64 cols. grid.x = ceil(M/256), grid.y = N/64.
// K % 64 == 0, N % 64 == 0.
//
// B panel is staged in LDS in *fragment-major* order: Bs[s][t][lane] (float2),
// where for K-step s and column-tile t, lane l = g*16+ml needs
// (B[kp+4s+2g][col0+16t+ml], B[kp+4s+2g+1][...]) contiguously. Every
// b-fragment read is then a single lane-linear ds_load_b64 at an immediate
// offset: no address math, no repacking movs, no bank conflicts.
__global__ __launch_bounds__(256)
void rgcn_wmma_gemm(const float* __restrict__ A, const float* __restrict__ B,
                    const float* __restrict__ bias, float* __restrict__ C,
                    int M, int K, int N)
{
    __shared__ float Bsf[(TILE_K / 4) * 4 * 32 * 2];   // 4096 floats = 16 KB

    const int lane = threadIdx.x & 31;
    const int wave = threadIdx.x >> 5;
    const int g    = lane >> 4;     // half-wave group (0/1)
    const int ml   = lane & 15;
    const int row0 = blockIdx.x * 256 + wave * 32;
    const int col0 = blockIdx.y * 64;

    // Clamp A rows so out-of-range rows read valid memory (stores guarded).
    int r0c = row0 + ml;      if (r0c > M - 1) r0c = M - 1;
    int r1c = row0 + 16 + ml; if (r1c > M - 1) r1c = M - 1;
    const float* __restrict__ Ap0 = A + (long long)r0c * K + (g << 1);
    const float* __restrict__ Ap1 = A + (long long)r1c * K + (g << 1);

    v8f acc0[4] = {v8f{}, v8f{}, v8f{}, v8f{}};
    v8f acc1[4] = {v8f{}, v8f{}, v8f{}, v8f{}};

    for (int kp = 0; kp < K; kp += TILE_K) {
        // Cooperative fill: coalesced float4 global reads, swizzled scalar
        // stores into fragment-major layout.
        {
            const int r0 = threadIdx.x >> 4;        // 0..15
            const int cq = (threadIdx.x & 15) << 2; // 0,4,...,60
            #pragma unroll
            for (int rr = 0; rr < TILE_K; rr += 16) {
                const int r = r0 + rr;              // k within panel
                const float4 vv = *(const float4*)(B + (long long)(kp + r) * N + col0 + cq);
                const int s  = r >> 2;
                const int gb = (r & 3) >> 1;
                const int vb = r & 1;
                const float tmp[4] = {vv.x, vv.y, vv.z, vv.w};
                #pragma unroll
                for (int i = 0; i < 4; ++i) {
                    const int c = cq + i;
                    const int t = c >> 4;
                    const int mlb = c & 15;
                    Bsf[(((((s << 2) + t) << 5) + (gb << 4) + mlb) << 1) + vb] = tmp[i];
                }
            }
        }
        __syncthreads();

        #pragma unroll
        for (int s = 0; s < TILE_K / 4; ++s) {
            const v2f a0 = *(const v2f*)(Ap0 + kp + (s << 2));
            const v2f a1 = *(const v2f*)(Ap1 + kp + (s << 2));
            #pragma unroll
            for (int t = 0; t < 4; ++t) {
                const v2f b = *(const v2f*)&Bsf[((((s << 2) + t) << 5) + lane) << 1];
                acc0[t] = wmma_f32(a0, b, acc0[t]);
                acc1[t] = wmma_f32(a1, b, acc1[t]);
            }
        }
        __syncthreads();
    }

    // C/D layout: VGPR v holds M = v + 8*g, N = ml.
    #pragma unroll
    for (int t = 0; t < 4; ++t) {
        const int n = col0 + t * 16 + ml;
        const float bv = bias ? bias[n] : 0.0f;
        #pragma unroll
        for (int v = 0; v < 8; ++v) {
            const int row = row0 + (g << 3) + v;
            if (row < M) C[(long long)row * N + n] = acc0[t][v] + bv;
            const int row1 = row + 16;
            if (row1 < M) C[(long long)row1 * N + n] = acc1[t][v] + bv;
        }
    }
}

__global__ void rgcn_deg_kernel(float* __restrict__ deg, const int* __restrict__ dst, int E)
{
    int e = blockIdx.x * blockDim.x + threadIdx.x;
    if (e < E) atomicAdd(&deg[dst[e]], 1.0f);
}

__global__ void rgcn_invdeg_kernel(float* __restrict__ deg, int n)
{
    int i = blockIdx.x * blockDim.x + threadIdx.x;
    if (i < n) deg[i] = 1.0f / fmaxf(deg[i], 1.0f);
}

// One thread per (edge, 4-float feature chunk).
// out[dst] += m[src] * invdeg[dst]  (scaled scatter: no per-relation accumulators)
__global__ void rgcn_scatter_kernel(float* __restrict__ out, const float* __restrict__ m,
                                    const int* __restrict__ src, const int* __restrict__ dst,
                                    const float* __restrict__ invdeg,
                                    long long total, int F, int shift)
{
    long long tid = blockIdx.x * (long long)blockDim.x + threadIdx.x;
    if (tid >= total) return;
    const int c       = (int)(tid & ((1 << shift) - 1));
    const long long e = tid >> shift;
    const int s = src[e];
    const int d = dst[e];
    const float w = invdeg[d];
    const float4 v = *(const float4*)(m + (long long)s * F + (c << 2));
    float* o = out + (long long)d * F + (c << 2);
    atomicAdd(o + 0, v.x * w);
    atomicAdd(o + 1, v.y * w);
    atomicAdd(o + 2, v.z * w);
    atomicAdd(o + 3, v.w * w);
}

// h += sum_{r=0..2} b[r]; optional ReLU. F is a power of two.
__global__ void rgcn_bias_act_kernel(float* __restrict__ h, const float* __restrict__ b,
                                     long long total, int Fmask, int F, int relu)
{
    long long i = blockIdx.x * (long long)blockDim.x + threadIdx.x;
    if (i >= total) return;
    const int j = (int)(i & Fmask);
    float s = h[i] + b[j] + b[F + j] + b[2 * F + j];
    if (relu) s = fmaxf(s, 0.0f);
    h[i] = s;
}

static inline unsigned cdiv(long long a, long long b) { return (unsigned)((a + b - 1) / b); }

extern "C" void kernel_launch(void* const* d_in, const int* in_sizes, int n_in,
                              void* d_out, int out_size, void* d_ws, size_t ws_size,
                              hipStream_t stream)
{
    (void)in_sizes; (void)n_in; (void)out_size; (void)ws_size;

    const float* x   = (const float*)d_in[0];   // 200000 x 128
    const float* W0  = (const float*)d_in[1];   // 3 x 128 x 256
    const float* b0  = (const float*)d_in[2];   // 3 x 256
    const float* W1  = (const float*)d_in[3];   // 3 x 256 x 256
    const float* b1  = (const float*)d_in[4];   // 3 x 256
    const float* W2  = (const float*)d_in[5];   // 3 x 256 x 128
    const float* b2  = (const float*)d_in[6];   // 3 x 128
    const float* Wsl = (const float*)d_in[7];   // 256 x 128
    const float* bsl = (const float*)d_in[8];   // 128
    const int* e0s = (const int*)d_in[9];
    const int* e0d = (const int*)d_in[10];
    const int* e1s = (const int*)d_in[11];
    const int* e1d = (const int*)d_in[12];
    const int* e2s = (const int*)d_in[13];
    const int* e2d = (const int*)d_in[14];
    float* out = (float*)d_out;                 // 25000 x 128

    const int N0 = 200000, N1 = 100000, N2 = 50000, N3 = 25000;
    const int E0 = 800000, E1 = 400000, E2 = 200000;
    const int R = 3;

    // Workspace carve-up (floats): m | h1 | h2 | deg
    float* m   = (float*)d_ws;                       // up to 200000*256
    float* h1  = m  + (long long)N0 * 256;           // 100000*256
    float* h2  = h1 + (long long)N1 * 256;           // 50000*256
    float* deg = h2 + (long long)N2 * 256;           // up to 100000

    // One relational-conv layer: for each relation r,
    //   m = hin @ W[r]; invdeg = 1/max(deg_r,1); hout[dst] += m[src]*invdeg[dst]
    auto layer = [&](const float* hin, float* hout, const float* W,
                     int Ms, int Kd, int Nd, const int* esrc, const int* edst,
                     int E, int Ndst) {
        const int shift = (Nd == 256) ? 6 : 5;  // Nd/4 chunks per edge
        for (int r = 0; r < R; ++r) {
            dim3 gg(cdiv(Ms, 256), (unsigned)(Nd / 64));
            rgcn_wmma_gemm<<<gg, 256, 0, stream>>>(
                hin, W + (long long)r * Kd * Nd, nullptr, m, Ms, Kd, Nd);

            hipMemsetAsync(deg, 0, (size_t)Ndst * sizeof(float), stream);
            rgcn_deg_kernel<<<cdiv(E, 256), 256, 0, stream>>>(deg, edst + (long long)r * E, E);
            rgcn_invdeg_kernel<<<cdiv(Ndst, 256), 256, 0, stream>>>(deg, Ndst);

            const long long total = (long long)E << shift;
            rgcn_scatter_kernel<<<cdiv(total, 256), 256, 0, stream>>>(
                hout, m, esrc + (long long)r * E, edst + (long long)r * E,
                deg, total, Nd, shift);
        }
    };

    // Layer 0: x(200000x128) -> h1(100000x256), ReLU
    hipMemsetAsync(h1, 0, (size_t)N1 * 256 * sizeof(float), stream);
    layer(x, h1, W0, N0, 128, 256, e0s, e0d, E0, N1);
    rgcn_bias_act_kernel<<<cdiv((long long)N1 * 256, 256), 256, 0, stream>>>(
        h1, b0, (long long)N1 * 256, 255, 256, 1);

    // Layer 1: h1(100000x256) -> h2(50000x256), ReLU
    hipMemsetAsync(h2, 0, (size_t)N2 * 256 * sizeof(float), stream);
    layer(h1, h2, W1, N1, 256, 256, e1s, e1d, E1, N2);
    rgcn_bias_act_kernel<<<cdiv((long long)N2 * 256, 256), 256, 0, stream>>>(
        h2, b1, (long long)N2 * 256, 255, 256, 1);

    // Skip connection: out = h2[:25000] @ W_sl + b_sl   (writes all of out)
    {
        dim3 gg(cdiv(N3, 256), 2);
        rgcn_wmma_gemm<<<gg, 256, 0, stream>>>(h2, Wsl, bsl, out, N3, 256, 128);
    }

    // Layer 2: h2(50000x256) -> accumulate into out(25000x128), no ReLU
    layer(h2, out, W2, N2, 256, 128, e2s, e2d, E2, N3);
    rgcn_bias_act_kernel<<<cdiv((long long)N3 * 128, 256), 256, 0, stream>>>(
        out, b2, (long long)N3 * 128, 127, 128, 0);
}